// PAM_Module_56332791054531
// MI455X (gfx1250) — compile-verified
//
#include <hip/hip_runtime.h>
#include <hip/hip_bf16.h>
#include <math.h>

// PAM (position attention) for B=2, C=64, D=H=W=16 -> N = 4096.
// Phase 1: Q/K/V = (dw ⊙ x) mixed by pw (64x64 matmul), stored [b][n][c].
// Phase 2: flash-attention, transposed formulation:
//   S^T = K·Q^T  (rows j in VGPRs, query index i across lanes)
//   per-lane softmax stats, O^T = V^T·P with per-lane rescale,
//   coalesced epilogue into [b][c][i].
// K/V j-tiles staged double-buffered via GLOBAL_LOAD_ASYNC_TO_LDS_B128
// (ASYNCcnt-tracked DMA), matmuls via V_WMMA_F32_16X16X4_F32 (fp32 precision).

typedef float v2f __attribute__((ext_vector_type(2)));
typedef float v8f __attribute__((ext_vector_type(8)));

#define BATCH 2
#define CH    64
#define NPOS  4096
#define JTILES (NPOS / 16)

#define KS_STRIDE 68   // row pad: fragment addresses hit banks 4m+c -> clean
#define PS_STRIDE 18

static __device__ __forceinline__ v8f wmma_f32_16x16x4(v2f a, v2f b, v8f c) {
  // (neg_a, A, neg_b, B, c_mod, C, reuse_a, reuse_b)
  return __builtin_amdgcn_wmma_f32_16x16x4_f32(false, a, false, b, (short)0, c,
                                               false, false);
}

// Async DMA: 16B global -> LDS, tracked by ASYNCcnt.  The LDS destination is
// the low 32 bits of the generic shared-space address (LDS aperture occupies
// ADDR[63:32]).
static __device__ __forceinline__ void async_copy_b128(void* lds_dst,
                                                       const void* gsrc) {
  const unsigned int       ldsa = (unsigned int)(unsigned long long)lds_dst;
  const unsigned long long ga   = (unsigned long long)gsrc;
  __asm__ volatile("global_load_async_to_lds_b128 %0, %1, off"
                   :
                   : "v"(ldsa), "v"(ga)
                   : "memory");
}
static __device__ __forceinline__ void wait_async0() {
  __asm__ volatile("s_wait_asynccnt 0" ::: "memory");
}

// ---------------------------------------------------------------------------
// Phase 1: QKV projection. Block = 256 threads, one (b, 64-n tile), all three
// projections. Output layout [b][n][c] so attention tiles are contiguous 4KB.
// ---------------------------------------------------------------------------
__global__ __launch_bounds__(256) void qkv_proj_kernel(
    const float* __restrict__ x,
    const float* __restrict__ qdw, const float* __restrict__ qpw,
    const float* __restrict__ kdw, const float* __restrict__ kpw,
    const float* __restrict__ vdw, const float* __restrict__ vpw,
    float* __restrict__ Qo, float* __restrict__ Ko, float* __restrict__ Vo) {
  __shared__ float xs[CH * 65];  // x tile [c][n], padded
  __shared__ float wl[CH * 65];  // weights [o][c], padded, dw folded in

  const int tid = threadIdx.x;
  const int b   = blockIdx.x / (NPOS / 64);
  const int n0  = (blockIdx.x % (NPOS / 64)) * 64;

  {  // stage x tile, coalesced along n
    const int nl = tid & 63;
    const int c0 = tid >> 6;  // 0..3
#pragma unroll
    for (int cc = 0; cc < CH; cc += 4) {
      const int c = cc + c0;
      xs[c * 65 + nl] = x[((size_t)b * CH + c) * NPOS + n0 + nl];
    }
  }

  const float* dws[3]  = {qdw, kdw, vdw};
  const float* pws[3]  = {qpw, kpw, vpw};
  float*       outs[3] = {Qo, Ko, Vo};

  const int o    = tid & 63;
  const int nsub = tid >> 6;  // 0..3 -> 16 consecutive n each

#pragma unroll 1
  for (int mtx = 0; mtx < 3; ++mtx) {
    __syncthreads();  // xs ready (iter 0) / previous compute done with wl
    const float* pw = pws[mtx];
    const float* dw = dws[mtx];
#pragma unroll
    for (int p = 0; p < CH * CH; p += 256) {
      const int i = p + tid;
      wl[(i >> 6) * 65 + (i & 63)] = pw[i] * dw[i & 63];
    }
    __syncthreads();

    float acc[16];
#pragma unroll
    for (int ni = 0; ni < 16; ++ni) acc[ni] = 0.f;
    for (int c = 0; c < CH; ++c) {
      const float w = wl[o * 65 + c];
#pragma unroll
      for (int ni = 0; ni < 16; ++ni)
        acc[ni] = fmaf(w, xs[c * 65 + nsub * 16 + ni], acc[ni]);
    }
    float* outp = outs[mtx];
#pragma unroll
    for (int ni = 0; ni < 16; ++ni) {
      const int n = nsub * 16 + ni;
      outp[((size_t)b * NPOS + n0 + n) * CH + o] = acc[ni];  // coalesced in o
    }
  }
}

// ---------------------------------------------------------------------------
// Phase 2: fused attention (transposed). Block = 128 threads (4 waves), each
// wave owns a 16-query tile; the K/V j-tile is double-buffered in LDS and
// filled by async global->LDS DMA while the WMMAs run.
// ---------------------------------------------------------------------------
__global__ __launch_bounds__(128) void pam_attn_kernel(
    const float* __restrict__ Q, const float* __restrict__ K,
    const float* __restrict__ V, const float* __restrict__ x,
    const float* __restrict__ gamma, float* __restrict__ out) {
  __shared__ float Ks[2][16 * KS_STRIDE];  // K tile [j][c], row-major padded
  __shared__ float Vs[2][16 * KS_STRIDE];  // V tile [j][c], row-major padded
  __shared__ float Ps[4][16 * PS_STRIDE];  // per-wave P^T relayout [i][j]

  const int tid  = threadIdx.x;
  const int wave = tid >> 5;
  const int lane = tid & 31;
  const int m    = lane & 15;  // query col i (S^T/D), row c (V^T A-frag)
  const int h    = lane >> 4;  // half-wave: K-offset select / row+8

  const int b      = blockIdx.x / (NPOS / 64);
  const int i_base = (blockIdx.x % (NPOS / 64)) * 64 + wave * 16;

  // Preload this wave's Q as 16 B-fragments: b.x = Q[i][k0+2h], b.y = +1.
  v2f qb[16];
  {
    const float* Qr = Q + ((size_t)b * NPOS + i_base + m) * CH + 2 * h;
#pragma unroll
    for (int kk = 0; kk < 16; ++kk) qb[kk] = *(const v2f*)(Qr + kk * 4);
  }

  v8f oacc[4];  // O^T fragments: rows c = g*16 + v + 8h, cols i across lanes
#pragma unroll
  for (int g = 0; g < 4; ++g)
    oacc[g] = (v8f){0.f, 0.f, 0.f, 0.f, 0.f, 0.f, 0.f, 0.f};
  float mrun = -3.0e38f;  // per-lane: running max for column i
  float lrun = 0.f;       // per-lane: running sum for column i

  const float* Kb = K + (size_t)b * NPOS * CH;
  const float* Vb = V + (size_t)b * NPOS * CH;

  // Each thread DMAs 2x16B of K and 2x16B of V per tile (tile = 1024 floats).
  const int j0 = tid >> 4;        // 0..7
  const int c4 = (tid & 15) * 4;  // 0..60

  auto stage_async = [&](int jt, int buf) {
    const float* Kt = Kb + (size_t)jt * 16 * CH;
    const float* Vt = Vb + (size_t)jt * 16 * CH;
    async_copy_b128(&Ks[buf][j0 * KS_STRIDE + c4],       Kt + j0 * CH + c4);
    async_copy_b128(&Ks[buf][(j0 + 8) * KS_STRIDE + c4], Kt + (j0 + 8) * CH + c4);
    async_copy_b128(&Vs[buf][j0 * KS_STRIDE + c4],       Vt + j0 * CH + c4);
    async_copy_b128(&Vs[buf][(j0 + 8) * KS_STRIDE + c4], Vt + (j0 + 8) * CH + c4);
  };

  stage_async(0, 0);

#pragma unroll 1
  for (int jt = 0; jt < JTILES; ++jt) {
    const int cur = jt & 1;
    wait_async0();    // this wave's DMA for buf[cur] has landed in LDS
    __syncthreads();  // all waves' DMA done; buf[cur^1] no longer being read

    const bool more = (jt + 1 < JTILES);
    if (more) stage_async(jt + 1, cur ^ 1);  // DMA overlaps the WMMA section

    // ---- S^T = K * Q^T : A = K tile (M=j, K=c), B = Q^T ----
    const float* ks = &Ks[cur][m * KS_STRIDE + 2 * h];
    v8f st = (v8f){0.f, 0.f, 0.f, 0.f, 0.f, 0.f, 0.f, 0.f};
#pragma unroll
    for (int kk = 0; kk < 16; ++kk) {
      const v2f ka = *(const v2f*)(ks + kk * 4);
      st = wmma_f32_16x16x4(ka, qb[kk], st);
    }

    // ---- per-lane online softmax for query column i = m ----
    float tmax = st[0];
#pragma unroll
    for (int v = 1; v < 8; ++v) tmax = fmaxf(tmax, st[v]);
    tmax = fmaxf(tmax, __shfl_xor(tmax, 16, 32));  // merge half-waves
    const float mn    = fmaxf(mrun, tmax);
    const float scale = __expf(mrun - mn);
    mrun              = mn;
    float psum        = 0.f;
#pragma unroll
    for (int v = 0; v < 8; ++v) {
      const float p = __expf(st[v] - mn);
      st[v]         = p;
      psum += p;
    }
    psum += __shfl_xor(psum, 16, 32);
    lrun = lrun * scale + psum;
#pragma unroll
    for (int g = 0; g < 4; ++g) {
#pragma unroll
      for (int v = 0; v < 8; ++v) oacc[g][v] *= scale;  // per-lane scalar
    }

    // ---- relayout P^T (D-layout -> B-layout) via per-wave LDS ----
#pragma unroll
    for (int v = 0; v < 8; ++v)
      Ps[wave][m * PS_STRIDE + v + 8 * h] = st[v];
    __asm__ volatile("s_wait_dscnt 0" ::: "memory");  // same-wave DS ordering

    // ---- O^T += V^T * P : A = V^T (M=c, K=j), B = P (K=j, N=i) ----
    const float* vs = Vs[cur];
#pragma unroll
    for (int kk = 0; kk < 4; ++kk) {
      const int jj = kk * 4 + 2 * h;
      const v2f pb = *(const v2f*)(&Ps[wave][m * PS_STRIDE + kk * 4 + 2 * h]);
#pragma unroll
      for (int g = 0; g < 4; ++g) {
        v2f va;  // lanes read c = g*16+m -> consecutive -> bank-clean
        va.x = vs[jj * KS_STRIDE + g * 16 + m];
        va.y = vs[(jj + 1) * KS_STRIDE + g * 16 + m];
        oacc[g] = wmma_f32_16x16x4(va, pb, oacc[g]);
      }
    }
  }

  // ---- epilogue: out[b][c][i] = gamma*O[i][c]/l[i] + x[b][c][i] ----
  const float gm  = gamma[0];
  const float inv = 1.0f / lrun;  // per-lane (column i)
  const int   i   = i_base + m;
#pragma unroll
  for (int g = 0; g < 4; ++g) {
#pragma unroll
    for (int v = 0; v < 8; ++v) {
      const int    c   = g * 16 + v + 8 * h;
      const size_t idx = ((size_t)b * CH + c) * NPOS + i;
      out[idx] = gm * (oacc[g][v] * inv) + x[idx];  // coalesced in i
    }
  }
}

extern "C" void kernel_launch(void* const* d_in, const int* in_sizes, int n_in,
                              void* d_out, int out_size, void* d_ws,
                              size_t ws_size, hipStream_t stream) {
  (void)in_sizes; (void)n_in; (void)out_size; (void)ws_size;
  const float* x     = (const float*)d_in[0];
  const float* q_dw  = (const float*)d_in[1];
  const float* q_pw  = (const float*)d_in[2];
  const float* k_dw  = (const float*)d_in[3];
  const float* k_pw  = (const float*)d_in[4];
  const float* v_dw  = (const float*)d_in[5];
  const float* v_pw  = (const float*)d_in[6];
  const float* gamma = (const float*)d_in[7];
  float* out = (float*)d_out;

  float* Qw = (float*)d_ws;                    // [B][N][C]
  float* Kw = Qw + (size_t)BATCH * NPOS * CH;
  float* Vw = Kw + (size_t)BATCH * NPOS * CH;  // 6 MB scratch total

  qkv_proj_kernel<<<BATCH * (NPOS / 64), 256, 0, stream>>>(
      x, q_dw, q_pw, k_dw, k_pw, v_dw, v_pw, Qw, Kw, Vw);
  pam_attn_kernel<<<BATCH * (NPOS / 64), 128, 0, stream>>>(
      Qw, Kw, Vw, x, gamma, out);
}